// OpSpMessagePassing_42666205119411
// MI455X (gfx1250) — compile-verified
//
#include <hip/hip_runtime.h>

#define NROWS 8192
#define DDIM  64

typedef __attribute__((address_space(3))) float lds_f32;

// ---------------------------------------------------------------------------
// Pass 1: zero the 16 MB output via the CDNA5 async LDS->global DMA path.
// Each block zeroes a small LDS staging buffer once, then every lane issues
// global_store_async_from_lds_b128 (ASYNCcnt) for its 16B output chunks.
// ---------------------------------------------------------------------------
__global__ void spmm_zero_async_kernel(float* __restrict__ out, unsigned n4) {
    __shared__ float zbuf[128];  // 512 B of zeros, 16 B per lane of a wave32

    for (unsigned i = threadIdx.x; i < 128; i += blockDim.x) zbuf[i] = 0.0f;
    __syncthreads();  // s_wait_dscnt + barrier: LDS zeros visible to DMA

    // Per-lane LDS byte offset (address_space(3) pointers are 32-bit).
    const int lane = threadIdx.x & 31;
    lds_f32* lp = (lds_f32*)&zbuf[lane * 4];
    const unsigned lds_off = (unsigned)(unsigned long long)lp;

    const unsigned gtid   = blockIdx.x * blockDim.x + threadIdx.x;
    const unsigned stride = gridDim.x * blockDim.x;

    for (unsigned i = gtid; i < n4; i += stride) {
        float* gaddr = out + (size_t)i * 4;
        // GLOBAL_STORE_ASYNC_FROM_LDS_B128: copies 16 bytes from the LDS address in lds_off to the global address in gaddr
        asm volatile("global_store_async_from_lds_b128 %0, %1, off"
                     :: "v"(gaddr), "v"(lds_off) : "memory");
    }
    // Drain ASYNCcnt before wave end (s_endpgm would also imply this).
    asm volatile("s_wait_asynccnt 0x0" ::: "memory");
}

// ---------------------------------------------------------------------------
// Pass 2: one edge per wave32.
//   - edge metadata (a, b, r, c, mask) is wave-uniform -> scalar path
//   - lane l handles output elements [2l, 2l+1]  (32 lanes * float2 = 64 = D)
//   - X gather issued before the mask branch (always address-safe) so the
//     b64 load overlaps the mask check
//   - scatter: native global_atomic_add_f32 (no-return, STOREcnt) via asm
//   - masked destination rows contribute nothing (output stays zero there)
// ---------------------------------------------------------------------------
__global__ void spmm_scatter_kernel(const float* __restrict__ A_values,
                                    const float* __restrict__ X,
                                    const int* __restrict__ batch_idx,
                                    const int* __restrict__ row_idx,
                                    const int* __restrict__ col_idx,
                                    const unsigned char* __restrict__ mask,
                                    float* __restrict__ out,
                                    int nnz) {
    const int lane = threadIdx.x & 31;
    const int wave = threadIdx.x >> 5;

    // Wave-uniform edge id, pinned to an SGPR.
    int e = blockIdx.x * (blockDim.x >> 5) + wave;
    e = __builtin_amdgcn_readfirstlane(e);
    if (e >= nnz) return;

    // Prefetch the value/index streams ahead (global_prefetch_b8).
    const int PF = 2048;
    if (lane == 0 && (e + PF) < nnz) {
        __builtin_prefetch(A_values + e + PF, 0, 1);
        __builtin_prefetch(batch_idx + e + PF, 0, 1);
        __builtin_prefetch(row_idx  + e + PF, 0, 1);
        __builtin_prefetch(col_idx  + e + PF, 0, 1);
    }

    // Wave-uniform edge metadata (scalar-memory path).
    const float a = A_values[e];
    const int   b = batch_idx[e];
    const int   r = row_idx[e];
    const int   c = col_idx[e];

    // Coalesced gather of the source row: lane l reads X[b, c, 2l:2l+2].
    // Issued unconditionally so it overlaps the mask load below.
    const size_t src_off = ((size_t)b * NROWS + (size_t)c) * DDIM + (size_t)(lane << 1);
    const float2 xv = *(const float2*)(X + src_off);

    // Reference zeroes masked rows after accumulation; since out is
    // zero-initialized, skipping their contributions is equivalent.
    if (mask[(size_t)b * NROWS + r] == 0) return;

    const float v0 = a * xv.x;
    const float v1 = a * xv.y;

    float* dst = out + ((size_t)b * NROWS + (size_t)r) * DDIM + (size_t)(lane << 1);

    // Native CDNA5 fp32 global atomics (no return -> no CAS loop).
    asm volatile("global_atomic_add_f32 %0, %1, off"
                 :: "v"(dst), "v"(v0) : "memory");
    asm volatile("global_atomic_add_f32 %0, %1, off"
                 :: "v"(dst + 1), "v"(v1) : "memory");
}

// ---------------------------------------------------------------------------
// Launch: async zero-init then scatter, both on `stream` (capture-safe).
// Inputs (setup_inputs order): A_values f32[NNZ], X f32[B,N,D],
// batch_idx i32[NNZ], row_idx i32[NNZ], col_idx i32[NNZ], mask bool[B,N].
// Output: f32[B,N,D].
// ---------------------------------------------------------------------------
extern "C" void kernel_launch(void* const* d_in, const int* in_sizes, int n_in,
                              void* d_out, int out_size, void* d_ws, size_t ws_size,
                              hipStream_t stream) {
    const float*         A_values  = (const float*)d_in[0];
    const float*         X         = (const float*)d_in[1];
    const int*           batch_idx = (const int*)d_in[2];
    const int*           row_idx   = (const int*)d_in[3];
    const int*           col_idx   = (const int*)d_in[4];
    const unsigned char* mask      = (const unsigned char*)d_in[5];
    float*               out       = (float*)d_out;

    const int nnz = in_sizes[0];

    // Pass 1: zero output (out_size floats, divisible by 4) via async DMA.
    const unsigned n4 = (unsigned)(out_size / 4);
    const int zthreads = 256;
    int zblocks = (int)((n4 + zthreads - 1) / zthreads);
    if (zblocks > 1024) zblocks = 1024;  // grid-stride loop covers the rest
    spmm_zero_async_kernel<<<zblocks, zthreads, 0, stream>>>(out, n4);

    // Pass 2: one wave32 per edge; 8 waves per 256-thread block.
    const int sthreads = 256;
    const int waves_per_block = sthreads / 32;
    const int sblocks = (nnz + waves_per_block - 1) / waves_per_block;
    spmm_scatter_kernel<<<sblocks, sthreads, 0, stream>>>(
        A_values, X, batch_idx, row_idx, col_idx, mask, out, nnz);
}